// SE3Transformer_81758997447169
// MI455X (gfx1250) — compile-verified
//
#include <hip/hip_runtime.h>
#include <math.h>

// ---------------- problem constants ----------------
#define NN   30000      // nodes
#define NE   960000     // edges
#define CF   32         // channels per degree
#define F0V  64         // input node features
#define HID  32         // radial hidden
#define D2   64         // 2*CF
// N and E are exact multiples of 16 (30000/16=1875, 960000/16=60000).

typedef __attribute__((ext_vector_type(2))) float v2f;
typedef __attribute__((ext_vector_type(8))) float v8f;

// ---------------- WMMA f32 16x16x4 helpers ----------------
__device__ __forceinline__ v8f wmma4(v2f a, v2f b, v8f c) {
  return __builtin_amdgcn_wmma_f32_16x16x4_f32(false, a, false, b, (short)0, c,
                                               false, false);
}
// A: 16xK row-major (lanes 0-15 rows M, half=lane>>4 selects K pair)
__device__ __forceinline__ v2f loadA(const float* A, int lda, int k0, int lane) {
  int m = lane & 15, kk = k0 + ((lane >> 4) << 1);
  v2f a; a.x = A[m * lda + kk]; a.y = A[m * lda + kk + 1]; return a;
}
// B: KxN row-major, tile of 16 cols starting at n0
__device__ __forceinline__ v2f loadB(const float* B, int ldb, int k0, int n0, int lane) {
  int n = n0 + (lane & 15), kk = k0 + ((lane >> 4) << 1);
  v2f b; b.x = B[kk * ldb + n]; b.y = B[(kk + 1) * ldb + n]; return b;
}
__device__ __forceinline__ v8f gemm_tile(const float* A, int lda,
                                         const float* B, int ldb,
                                         int K, int n0, int lane, v8f c) {
  for (int k0 = 0; k0 < K; k0 += 4)
    c = wmma4(loadA(A, lda, k0, lane), loadB(B, ldb, k0, n0, lane), c);
  return c;
}
__device__ __forceinline__ v8f zero_frag() {
  v8f c;
#pragma unroll
  for (int j = 0; j < 8; ++j) c[j] = 0.0f;
  return c;
}
__device__ __forceinline__ v8f bias_frag(const float* b, int n0, int lane) {
  float v = b[n0 + (lane & 15)];
  v8f c;
#pragma unroll
  for (int j = 0; j < 8; ++j) c[j] = v;
  return c;
}
// C/D element (j, lane) maps to row = j + 8*(lane>>4), col = n0 + (lane&15).

__device__ __forceinline__ void atomicMaxF(float* p, float v) {
  unsigned int* u = (unsigned int*)p;
  unsigned int old = *u;
  while (__uint_as_float(old) < v) {
    unsigned int assumed = old;
    old = atomicCAS(u, assumed, __float_as_uint(v));
    if (old == assumed) break;
  }
}
__device__ __forceinline__ float sigmoidf(float x) { return 1.0f / (1.0f + expf(-x)); }

// ---------------- node kernels ----------------
// h0 = elu(nf@W1+b1)@W2+b2   (one wave = 16 nodes)
__global__ void __launch_bounds__(32) k_init(const float* __restrict__ nf,
                                             const float* __restrict__ W1,
                                             const float* __restrict__ b1,
                                             const float* __restrict__ W2,
                                             const float* __restrict__ b2,
                                             float* __restrict__ h0) {
  __shared__ float ls_mid[16 * F0V];
  int lane = threadIdx.x, half = lane >> 4, ln = lane & 15;
  int n0 = blockIdx.x * 16;
  const float* A = nf + (size_t)n0 * F0V;
#pragma unroll
  for (int nt = 0; nt < 4; ++nt) {
    v8f c = bias_frag(b1, nt * 16, lane);
    c = gemm_tile(A, F0V, W1, F0V, F0V, nt * 16, lane, c);
#pragma unroll
    for (int j = 0; j < 8; ++j) {
      float x = c[j];
      x = x > 0.0f ? x : (expf(x) - 1.0f);  // ELU(alpha=1)
      ls_mid[(j + half * 8) * F0V + nt * 16 + ln] = x;
    }
  }
  __syncthreads();
#pragma unroll
  for (int nt = 0; nt < 2; ++nt) {
    v8f c = bias_frag(b2, nt * 16, lane);
    c = gemm_tile(ls_mid, F0V, W2, CF, F0V, nt * 16, lane, c);
#pragma unroll
    for (int j = 0; j < 8; ++j)
      h0[(size_t)(n0 + j + half * 8) * CF + nt * 16 + ln] = c[j];
  }
}

// h1[n,c,d] = x_l1[n,0,d] * Wl1[0,c]
__global__ void k_h1(const float* __restrict__ xl1, const float* __restrict__ Wl1,
                     float* __restrict__ h1) {
  int idx = blockIdx.x * blockDim.x + threadIdx.x;
  if (idx >= NN * CF) return;
  int n = idx >> 5, c = idx & 31;
  float w = Wl1[c];
#pragma unroll
  for (int d = 0; d < 3; ++d)
    h1[(size_t)n * 96 + c * 3 + d] = xl1[(size_t)n * 3 + d] * w;
}

__global__ void k_zero_layer(float* num0, float* num1, float* maxb, float* sumb) {
  int idx = blockIdx.x * blockDim.x + threadIdx.x;
  if (idx >= NN * 96) return;
  num1[idx] = 0.0f;
  if (idx < NN * CF) num0[idx] = 0.0f;
  if (idx < NN * 2) { maxb[idx] = -3.0e38f; sumb[idx] = 0.0f; }
}

// q = h0 @ Wq
__global__ void __launch_bounds__(32) k_q(const float* __restrict__ h0,
                                          const float* __restrict__ Wq,
                                          float* __restrict__ q) {
  int lane = threadIdx.x, half = lane >> 4, ln = lane & 15;
  int n0 = blockIdx.x * 16;
  const float* A = h0 + (size_t)n0 * CF;
#pragma unroll
  for (int nt = 0; nt < 2; ++nt) {
    v8f c = gemm_tile(A, CF, Wq, CF, CF, nt * 16, lane, zero_frag());
#pragma unroll
    for (int j = 0; j < 8; ++j)
      q[(size_t)(n0 + j + half * 8) * CF + nt * 16 + ln] = c[j];
  }
}

// ---------------- edge prologue (16 edges per wave) ----------------
__device__ __forceinline__ void edge_prologue(
    int e0, int lane, const int* esrc, const int* edst, const float* pos,
    const float* ef, const float* Wr1, const float* br1,
    int* s_src, int* s_dst, float* s_u, float* s_rin, float* s_hid) {
  int i = lane & 15;
  if (lane < 16) {
    int s = esrc[e0 + i], t = edst[e0 + i];
    s_src[i] = s; s_dst[i] = t;
    float rx = pos[t * 3 + 0] - pos[s * 3 + 0];
    float ry = pos[t * 3 + 1] - pos[s * 3 + 1];
    float rz = pos[t * 3 + 2] - pos[s * 3 + 2];
    float dd = sqrtf(rx * rx + ry * ry + rz * rz + 1e-8f);
    float inv = 1.0f / dd;
    s_u[i * 3 + 0] = rx * inv; s_u[i * 3 + 1] = ry * inv; s_u[i * 3 + 2] = rz * inv;
    s_rin[i * 3 + 0] = dd;
    s_rin[i * 3 + 1] = ef[(size_t)(e0 + i) * 2 + 0];
    s_rin[i * 3 + 2] = ef[(size_t)(e0 + i) * 2 + 1];
  }
  __syncthreads();
  float r0 = s_rin[i * 3 + 0], r1 = s_rin[i * 3 + 1], r2 = s_rin[i * 3 + 2];
  int kb = (lane >> 4) * 16;
#pragma unroll
  for (int kk = 0; kk < 16; ++kk) {
    int k = kb + kk;
    float a = br1[k] + r0 * Wr1[0 * HID + k] + r1 * Wr1[1 * HID + k] + r2 * Wr1[2 * HID + k];
    s_hid[i * HID + k] = a > 0.0f ? a : 0.0f;
  }
  __syncthreads();
}

// ---------------- attention edge pass 1: logits + segment max ----------------
__global__ void __launch_bounds__(128) k_edge1(
    const int* __restrict__ esrc, const int* __restrict__ edst,
    const float* __restrict__ pos, const float* __restrict__ ef,
    const float* __restrict__ h0, const float* __restrict__ h1,
    const float* __restrict__ q, const float* __restrict__ Wr1,
    const float* __restrict__ br1, const float* __restrict__ Wr2,
    const float* __restrict__ br2, const float* __restrict__ Wk,
    float* __restrict__ logit, float* __restrict__ maxb) {
  __shared__ int sh_src[4][16], sh_dst[4][16];
  __shared__ float sh_u[4][48], sh_rin[4][48], sh_hid[4][16 * HID], sh_m0[4][16 * CF];
  int wv = threadIdx.x >> 5, lane = threadIdx.x & 31;
  int e0 = (blockIdx.x * 4 + wv) * 16;
  edge_prologue(e0, lane, esrc, edst, pos, ef, Wr1, br1,
                sh_src[wv], sh_dst[wv], sh_u[wv], sh_rin[wv], sh_hid[wv]);
  int half = lane >> 4, ln = lane & 15;
  v8f w0[2], w2[2];  // radial paths 0 (cols 0..31) and 2 (cols 64..95)
#pragma unroll
  for (int nt = 0; nt < 2; ++nt) {
    w0[nt] = gemm_tile(sh_hid[wv], HID, Wr2, 128, HID, nt * 16, lane,
                       bias_frag(br2, nt * 16, lane));
    w2[nt] = gemm_tile(sh_hid[wv], HID, Wr2, 128, HID, 64 + nt * 16, lane,
                       bias_frag(br2, 64 + nt * 16, lane));
  }
#pragma unroll
  for (int nt = 0; nt < 2; ++nt) {
#pragma unroll
    for (int j = 0; j < 8; ++j) {
      int m = j + half * 8, c = nt * 16 + ln;
      int s = sh_src[wv][m];
      float h0v = h0[(size_t)s * CF + c];
      const float* hp = h1 + (size_t)s * 96 + c * 3;
      float dot = hp[0] * sh_u[wv][m * 3 + 0] + hp[1] * sh_u[wv][m * 3 + 1] +
                  hp[2] * sh_u[wv][m * 3 + 2];
      sh_m0[wv][m * CF + c] = w0[nt][j] * h0v + w2[nt][j] * dot;
    }
  }
  __syncthreads();
#pragma unroll
  for (int nt = 0; nt < 2; ++nt) {  // nt == head
    v8f kf = gemm_tile(sh_m0[wv], CF, Wk, CF, CF, nt * 16, lane, zero_frag());
#pragma unroll
    for (int j = 0; j < 8; ++j) {
      int m = j + half * 8, c = nt * 16 + ln;
      int t = sh_dst[wv][m];
      float p = kf[j] * q[(size_t)t * CF + c];
      p += __shfl_xor(p, 1); p += __shfl_xor(p, 2);
      p += __shfl_xor(p, 4); p += __shfl_xor(p, 8);
      if (ln == 0) {
        float lg = p * 0.25f;  // / sqrt(Dh=16)
        logit[(size_t)(e0 + m) * 2 + nt] = lg;
        atomicMaxF(&maxb[(size_t)t * 2 + nt], lg);
      }
    }
  }
}

// ---------------- attention edge pass 2: exp/sum + weighted scatter ----------
__global__ void __launch_bounds__(128) k_edge2(
    const int* __restrict__ esrc, const int* __restrict__ edst,
    const float* __restrict__ pos, const float* __restrict__ ef,
    const float* __restrict__ h0, const float* __restrict__ h1,
    const float* __restrict__ Wr1, const float* __restrict__ br1,
    const float* __restrict__ Wr2, const float* __restrict__ br2,
    const float* __restrict__ logit, const float* __restrict__ maxb,
    float* __restrict__ sumb, float* __restrict__ num0, float* __restrict__ num1) {
  __shared__ int sh_src[4][16], sh_dst[4][16];
  __shared__ float sh_u[4][48], sh_rin[4][48], sh_hid[4][16 * HID], sh_e[4][32];
  int wv = threadIdx.x >> 5, lane = threadIdx.x & 31;
  int e0 = (blockIdx.x * 4 + wv) * 16;
  edge_prologue(e0, lane, esrc, edst, pos, ef, Wr1, br1,
                sh_src[wv], sh_dst[wv], sh_u[wv], sh_rin[wv], sh_hid[wv]);
  {  // un-normalized attention weights + segment sum
    int i = lane & 15, hh = lane >> 4;
    int t = sh_dst[wv][i];
    float ev = expf(logit[(size_t)(e0 + i) * 2 + hh] - maxb[(size_t)t * 2 + hh]);
    sh_e[wv][i * 2 + hh] = ev;
    atomicAdd(&sumb[(size_t)t * 2 + hh], ev);
  }
  __syncthreads();
  int half = lane >> 4, ln = lane & 15;
  v8f wf[4][2];  // all 4 radial CG paths
#pragma unroll
  for (int p = 0; p < 4; ++p)
#pragma unroll
    for (int nt = 0; nt < 2; ++nt)
      wf[p][nt] = gemm_tile(sh_hid[wv], HID, Wr2, 128, HID, p * 32 + nt * 16, lane,
                            bias_frag(br2, p * 32 + nt * 16, lane));
#pragma unroll
  for (int nt = 0; nt < 2; ++nt) {
#pragma unroll
    for (int j = 0; j < 8; ++j) {
      int m = j + half * 8, c = nt * 16 + ln;
      int s = sh_src[wv][m], t = sh_dst[wv][m];
      float h0v = h0[(size_t)s * CF + c];
      const float* hp = h1 + (size_t)s * 96 + c * 3;
      float u0 = sh_u[wv][m * 3 + 0], u1 = sh_u[wv][m * 3 + 1], u2 = sh_u[wv][m * 3 + 2];
      float dot = hp[0] * u0 + hp[1] * u1 + hp[2] * u2;
      float eh = sh_e[wv][m * 2 + nt];  // head == c>>4 == nt
      float m0v = wf[0][nt][j] * h0v + wf[2][nt][j] * dot;
      atomicAdd(&num0[(size_t)t * CF + c], eh * m0v);
      float w1v = wf[1][nt][j], w3v = wf[3][nt][j];
      atomicAdd(&num1[(size_t)t * 96 + c * 3 + 0], eh * (w1v * h0v * u0 + w3v * hp[0]));
      atomicAdd(&num1[(size_t)t * 96 + c * 3 + 1], eh * (w1v * h0v * u1 + w3v * hp[1]));
      atomicAdd(&num1[(size_t)t * 96 + c * 3 + 2], eh * (w1v * h0v * u2 + w3v * hp[2]));
    }
  }
}

// ---------------- per-layer node epilogue: self-interaction + GNormSE3 ------
__global__ void __launch_bounds__(32) k_post(
    const float* __restrict__ num0, const float* __restrict__ num1,
    const float* __restrict__ sumb, const float* __restrict__ Ws0,
    const float* __restrict__ Ws1, const float* __restrict__ Wn0,
    const float* __restrict__ bn0, const float* __restrict__ Wn1,
    const float* __restrict__ bn1, float* __restrict__ h0, float* __restrict__ h1) {
  __shared__ float ls_h0[16 * CF], ls_o0[16 * CF], ls_o1[16 * CF * 3], ls_t[16 * CF];
  int lane = threadIdx.x, half = lane >> 4, ln = lane & 15;
  int n0 = blockIdx.x * 16;
  for (int idx = lane; idx < 16 * CF; idx += 32)
    ls_h0[idx] = h0[(size_t)(n0 + (idx >> 5)) * CF + (idx & 31)];
  __syncthreads();
#pragma unroll
  for (int nt = 0; nt < 2; ++nt) {  // out0 = num0/sum + h0@Ws0
    v8f c = gemm_tile(ls_h0, CF, Ws0, CF, CF, nt * 16, lane, zero_frag());
#pragma unroll
    for (int j = 0; j < 8; ++j) {
      int m = j + half * 8, cc = nt * 16 + ln, n = n0 + m;
      float den = sumb[(size_t)n * 2 + nt] + 1e-9f;
      ls_o0[m * CF + cc] = num0[(size_t)n * CF + cc] / den + c[j];
    }
  }
  for (int d = 0; d < 3; ++d) {  // out1_d = num1/sum + h1_d@Ws1
    __syncthreads();
    for (int idx = lane; idx < 16 * CF; idx += 32)
      ls_t[idx] = h1[(size_t)(n0 + (idx >> 5)) * 96 + (idx & 31) * 3 + d];
    __syncthreads();
#pragma unroll
    for (int nt = 0; nt < 2; ++nt) {
      v8f c = gemm_tile(ls_t, CF, Ws1, CF, CF, nt * 16, lane, zero_frag());
#pragma unroll
      for (int j = 0; j < 8; ++j) {
        int m = j + half * 8, cc = nt * 16 + ln, n = n0 + m;
        float den = sumb[(size_t)n * 2 + nt] + 1e-9f;
        ls_o1[(m * CF + cc) * 3 + d] = num1[(size_t)n * 96 + cc * 3 + d] / den + c[j];
      }
    }
  }
  __syncthreads();
  for (int idx = lane; idx < 16 * CF; idx += 32) ls_t[idx] = fabsf(ls_o0[idx]);
  __syncthreads();
#pragma unroll
  for (int nt = 0; nt < 2; ++nt) {  // h0 = sigmoid(|o0|@Wn0+bn0) * o0
    v8f c = gemm_tile(ls_t, CF, Wn0, CF, CF, nt * 16, lane, bias_frag(bn0, nt * 16, lane));
#pragma unroll
    for (int j = 0; j < 8; ++j) {
      int m = j + half * 8, cc = nt * 16 + ln;
      h0[(size_t)(n0 + m) * CF + cc] = sigmoidf(c[j]) * ls_o0[m * CF + cc];
    }
  }
  __syncthreads();
  for (int idx = lane; idx < 16 * CF; idx += 32) {
    float a = ls_o1[idx * 3 + 0], b = ls_o1[idx * 3 + 1], c2 = ls_o1[idx * 3 + 2];
    ls_t[idx] = sqrtf(a * a + b * b + c2 * c2 + 1e-8f);
  }
  __syncthreads();
#pragma unroll
  for (int nt = 0; nt < 2; ++nt) {  // h1 = sigmoid(n1@Wn1+bn1) * o1
    v8f c = gemm_tile(ls_t, CF, Wn1, CF, CF, nt * 16, lane, bias_frag(bn1, nt * 16, lane));
#pragma unroll
    for (int j = 0; j < 8; ++j) {
      int m = j + half * 8, cc = nt * 16 + ln;
      float s1 = sigmoidf(c[j]);
      size_t base = (size_t)(n0 + m) * 96 + cc * 3;
      h1[base + 0] = s1 * ls_o1[(m * CF + cc) * 3 + 0];
      h1[base + 1] = s1 * ls_o1[(m * CF + cc) * 3 + 1];
      h1[base + 2] = s1 * ls_o1[(m * CF + cc) * 3 + 2];
    }
  }
}

// ---------------- final conv (degree-1 output is dead -> skipped) -----------
__global__ void k_zero_final(float* a0, float* red) {
  int idx = blockIdx.x * blockDim.x + threadIdx.x;
  if (idx < NN * D2) a0[idx] = 0.0f;
  if (idx < 2) red[idx] = 0.0f;
}

__global__ void __launch_bounds__(128) k_edge3(
    const int* __restrict__ esrc, const int* __restrict__ edst,
    const float* __restrict__ pos, const float* __restrict__ ef,
    const float* __restrict__ h0, const float* __restrict__ h1,
    const float* __restrict__ Wr1, const float* __restrict__ br1,
    const float* __restrict__ Wr2, const float* __restrict__ br2,
    float* __restrict__ a0) {
  __shared__ int sh_src[4][16], sh_dst[4][16];
  __shared__ float sh_u[4][48], sh_rin[4][48], sh_hid[4][16 * HID];
  int wv = threadIdx.x >> 5, lane = threadIdx.x & 31;
  int e0 = (blockIdx.x * 4 + wv) * 16;
  edge_prologue(e0, lane, esrc, edst, pos, ef, Wr1, br1,
                sh_src[wv], sh_dst[wv], sh_u[wv], sh_rin[wv], sh_hid[wv]);
  int half = lane >> 4, ln = lane & 15;
#pragma unroll
  for (int nt = 0; nt < 2; ++nt) {
    v8f w0 = gemm_tile(sh_hid[wv], HID, Wr2, 128, HID, nt * 16, lane,
                       bias_frag(br2, nt * 16, lane));
    v8f w2 = gemm_tile(sh_hid[wv], HID, Wr2, 128, HID, 64 + nt * 16, lane,
                       bias_frag(br2, 64 + nt * 16, lane));
#pragma unroll
    for (int j = 0; j < 8; ++j) {
      int m = j + half * 8, c = nt * 16 + ln;
      int s = sh_src[wv][m], t = sh_dst[wv][m];
      float h0v = h0[(size_t)s * CF + c];
      const float* hp = h1 + (size_t)s * 96 + c * 3;
      float dot = hp[0] * sh_u[wv][m * 3 + 0] + hp[1] * sh_u[wv][m * 3 + 1] +
                  hp[2] * sh_u[wv][m * 3 + 2];
      atomicAdd(&a0[(size_t)t * D2 + c], w0[j] * h0v);        // m00
      atomicAdd(&a0[(size_t)t * D2 + 32 + c], w2[j] * dot);   // m10
    }
  }
}

__global__ void __launch_bounds__(32) k_finalnode(
    const float* __restrict__ a0, const float* __restrict__ h0,
    const float* __restrict__ Wmix0, const float* __restrict__ Wself0,
    float* __restrict__ hF) {
  int lane = threadIdx.x, half = lane >> 4, ln = lane & 15;
  int n0 = blockIdx.x * 16;
#pragma unroll
  for (int nt = 0; nt < 4; ++nt) {
    v8f c = zero_frag();
    c = gemm_tile(a0 + (size_t)n0 * D2, D2, Wmix0, D2, D2, nt * 16, lane, c);
    c = gemm_tile(h0 + (size_t)n0 * CF, CF, Wself0, D2, CF, nt * 16, lane, c);
#pragma unroll
    for (int j = 0; j < 8; ++j)
      hF[(size_t)(n0 + j + half * 8) * D2 + nt * 16 + ln] = c[j];
  }
}

// ---------------- head: (w.T/N) @ cat accumulated into red[2] ---------------
__global__ void __launch_bounds__(32) k_head(
    const float* __restrict__ hF, const float* __restrict__ Wc1,
    const float* __restrict__ bc1, const float* __restrict__ Wc2,
    const float* __restrict__ bc2, const float* __restrict__ Ww1,
    const float* __restrict__ bw1, const float* __restrict__ Ww2,
    const float* __restrict__ bw2, float* __restrict__ red) {
  __shared__ float ls_t1[16 * D2], ls_t2[16 * D2];
  int lane = threadIdx.x, half = lane >> 4, ln = lane & 15;
  int n0 = blockIdx.x * 16;
  const float* A = hF + (size_t)n0 * D2;
#pragma unroll
  for (int nt = 0; nt < 4; ++nt) {
    v8f c = gemm_tile(A, D2, Wc1, D2, D2, nt * 16, lane, bias_frag(bc1, nt * 16, lane));
    v8f w = gemm_tile(A, D2, Ww1, D2, D2, nt * 16, lane, bias_frag(bw1, nt * 16, lane));
#pragma unroll
    for (int j = 0; j < 8; ++j) {
      int m = j + half * 8, cc = nt * 16 + ln;
      ls_t1[m * D2 + cc] = fmaxf(c[j], 0.0f);
      ls_t2[m * D2 + cc] = fmaxf(w[j], 0.0f);
    }
  }
  __syncthreads();
  if (lane < 16) {
    float c0 = bc2[0], c1 = bc2[1], wv = bw2[0];
    for (int k = 0; k < D2; ++k) {
      float t = ls_t1[lane * D2 + k];
      c0 += t * Wc2[k * 2 + 0];
      c1 += t * Wc2[k * 2 + 1];
      wv += ls_t2[lane * D2 + k] * Ww2[k];
    }
    float s = wv * (1.0f / (float)NN);
    atomicAdd(&red[0], s * c0);
    atomicAdd(&red[1], s * c1);
  }
}

__global__ void k_softmax2(const float* __restrict__ red, float* __restrict__ out) {
  if (threadIdx.x == 0 && blockIdx.x == 0) {
    float a = red[0], b = red[1], m = fmaxf(a, b);
    float ea = expf(a - m), eb = expf(b - m), inv = 1.0f / (ea + eb);
    out[0] = ea * inv; out[1] = eb * inv;
  }
}

// ---------------- launcher ----------------
// d_in layout (insertion order of setup_inputs() / _init_params()):
//  0 node_feat[N,64] 1 x_l1[N,1,3] 2 pos[N,3] 3 edge_feat[E,2] 4 edge_src 5 edge_dst
//  6 W1[64,64] 7 b1[64] 8 W2[64,32] 9 b2[32] 10 Wl1[1,32]
//  11+12*l: Wr1[3,32] br1[32] Wr2[32,128] br2[128] Wq[32,32] Wk[32,32]
//           Ws0[32,32] Ws1[32,32] Wn0[32,32] bn0[32] Wn1[32,32] bn1[32]   (l=0,1)
//  35 Wr1f 36 br1f 37 Wr2f 38 br2f 39 Wmix0[64,64] 40 Wmix1 41 Wself0[32,64] 42 Wself1
//  43 Wc1[64,64] 44 bc1 45 Wc2[64,2] 46 bc2 47 Ww1 48 bw1 49 Ww2[64,1] 50 bw2
extern "C" void kernel_launch(void* const* d_in, const int* in_sizes, int n_in,
                              void* d_out, int out_size, void* d_ws, size_t ws_size,
                              hipStream_t stream) {
  (void)in_sizes; (void)n_in; (void)out_size; (void)ws_size;
  auto F = [&](int i) { return (const float*)d_in[i]; };
  const float* nf  = F(0);
  const float* xl1 = F(1);
  const float* pos = F(2);
  const float* ef  = F(3);
  const int* esrc = (const int*)d_in[4];
  const int* edst = (const int*)d_in[5];

  float* ws    = (float*)d_ws;
  float* h0    = ws;                 // N*32
  float* h1    = ws + 960000;        // N*96
  float* q     = ws + 3840000;       // N*32
  float* maxb  = ws + 4800000;       // N*2
  float* sumb  = ws + 4860000;       // N*2
  float* num0  = ws + 4920000;       // N*32   (num0+num1 reused as a0[N*64])
  float* num1  = ws + 5880000;       // N*96
  float* logit = ws + 8760000;       // E*2
  float* hF    = ws + 10680000;      // N*64
  float* red   = ws + 12600000;      // 2
  float* a0    = num0;

  k_init<<<NN / 16, 32, 0, stream>>>(nf, F(6), F(7), F(8), F(9), h0);
  k_h1<<<(NN * CF) / 256, 256, 0, stream>>>(xl1, F(10), h1);

  for (int l = 0; l < 2; ++l) {
    int b = 11 + 12 * l;
    const float *Wr1 = F(b), *br1 = F(b + 1), *Wr2 = F(b + 2), *br2 = F(b + 3);
    const float *Wq = F(b + 4), *Wk = F(b + 5), *Ws0 = F(b + 6), *Ws1 = F(b + 7);
    const float *Wn0 = F(b + 8), *bn0 = F(b + 9), *Wn1 = F(b + 10), *bn1 = F(b + 11);
    k_zero_layer<<<(NN * 96) / 256, 256, 0, stream>>>(num0, num1, maxb, sumb);
    k_q<<<NN / 16, 32, 0, stream>>>(h0, Wq, q);
    k_edge1<<<NE / 64, 128, 0, stream>>>(esrc, edst, pos, ef, h0, h1, q,
                                         Wr1, br1, Wr2, br2, Wk, logit, maxb);
    k_edge2<<<NE / 64, 128, 0, stream>>>(esrc, edst, pos, ef, h0, h1,
                                         Wr1, br1, Wr2, br2, logit, maxb,
                                         sumb, num0, num1);
    k_post<<<NN / 16, 32, 0, stream>>>(num0, num1, sumb, Ws0, Ws1,
                                       Wn0, bn0, Wn1, bn1, h0, h1);
  }

  k_zero_final<<<(NN * D2) / 256, 256, 0, stream>>>(a0, red);
  k_edge3<<<NE / 64, 128, 0, stream>>>(esrc, edst, pos, ef, h0, h1,
                                       F(35), F(36), F(37), F(38), a0);
  k_finalnode<<<NN / 16, 32, 0, stream>>>(a0, h0, F(39), F(41), hF);
  k_head<<<NN / 16, 32, 0, stream>>>(hF, F(43), F(44), F(45), F(46),
                                     F(47), F(48), F(49), F(50), red);
  k_softmax2<<<1, 32, 0, stream>>>(red, (float*)d_out);
}